// Model_16552803959180
// MI455X (gfx1250) — compile-verified
//
#include <hip/hip_runtime.h>
#include <hip/hip_bf16.h>

// Problem constants (match reference)
#define NUSERS   100000
#define NITEMS   200000
#define NNODES   (NUSERS + NITEMS)      // 300000
#define DIM      128
#define EIGD     32
#define NEDGES   600000
#define NLAYERS  2
#define NPATH    6

typedef __attribute__((ext_vector_type(2))) float v2f;
typedef __attribute__((ext_vector_type(8))) float v8f;

// ---------- helpers ----------

// Monotone float -> u32 key (order-preserving); memset(0) is identity for max.
__device__ __forceinline__ unsigned f2key(float f) {
  unsigned u = __float_as_uint(f);
  return (u & 0x80000000u) ? ~u : (u | 0x80000000u);
}
__device__ __forceinline__ float key2f(unsigned u) {
  return __uint_as_float((u & 0x80000000u) ? (u & 0x7fffffffu) : ~u);
}

__device__ __forceinline__ float wave_sum32(float v) {
#pragma unroll
  for (int m = 16; m > 0; m >>= 1) v += __shfl_xor(v, m, 32);
  return v;
}

// ---------- kernels ----------

// x = concat(user_emb, item_emb); out = x / 3   (float4 streaming)
__global__ void k_init(const float* __restrict__ ue, const float* __restrict__ ie,
                       float* __restrict__ x, float* __restrict__ out) {
  size_t i4 = (size_t)blockIdx.x * blockDim.x + threadIdx.x;
  const size_t tot4 = (size_t)NNODES * DIM / 4;
  if (i4 >= tot4) return;
  const size_t u4 = (size_t)NUSERS * DIM / 4;
  float4 v = (i4 < u4) ? ((const float4*)ue)[i4] : ((const float4*)ie)[i4 - u4];
  ((float4*)x)[i4] = v;
  float4 o;
  o.x = v.x * (1.0f / 3.0f); o.y = v.y * (1.0f / 3.0f);
  o.z = v.z * (1.0f / 3.0f); o.w = v.w * (1.0f / 3.0f);
  ((float4*)out)[i4] = o;
}

// In-place LayerNorm (no affine), one row per wave32, float4 per lane.
__global__ void k_layernorm(float* __restrict__ x) {
  int wid  = (int)((blockIdx.x * (size_t)blockDim.x + threadIdx.x) >> 5);
  int lane = threadIdx.x & 31;
  if (wid >= NNODES) return;
  float4* row = (float4*)(x + (size_t)wid * DIM);
  float4 v = row[lane];
  float mean = wave_sum32(v.x + v.y + v.z + v.w) * (1.0f / DIM);
  float dx = v.x - mean, dy = v.y - mean, dz = v.z - mean, dw = v.w - mean;
  float var = wave_sum32(dx * dx + dy * dy + dz * dz + dw * dw) * (1.0f / DIM);
  float rs = rsqrtf(var + 1e-5f);
  float4 o; o.x = dx * rs; o.y = dy * rs; o.z = dz * rs; o.w = dw * rs;
  row[lane] = o;
}

// Edge scores via f32 WMMA: one wave computes 16 edge dot-products as the
// diagonal of a 16x16 Gram tile, chaining V_WMMA_F32_16X16X4_F32 over
// K = DIM (A pre-scaled by 1/sqrt(D)) and K = EIGD (A pre-scaled by e^lambda0).
// Then writes s0/s1 and updates per-destination segment maxima.
__global__ void k_edge_scores(const float* __restrict__ y,
                              const float* __restrict__ eigs,
                              const int* __restrict__ i0,
                              const int* __restrict__ i1,
                              const int* __restrict__ pt,
                              const float* __restrict__ lambda0,
                              const float* __restrict__ pathw,
                              float* __restrict__ s0,
                              float* __restrict__ s1,
                              unsigned* __restrict__ m0,
                              unsigned* __restrict__ m1) {
  int wid  = (int)((blockIdx.x * (size_t)blockDim.x + threadIdx.x) >> 5);
  int lane = threadIdx.x & 31;
  long base = (long)wid * 16;
  if (base >= NEDGES) return;

  const float rsD   = 0.08838834764831845f;   // 1/sqrt(128)
  const float expl0 = __expf(lambda0[0]);

  // A-matrix 16x4 f32 layout: lanes 0-15 rows M=0..15 hold K={k0,k0+1},
  // lanes 16-31 hold K={k0+2,k0+3}. B (4x16) mirrored: lane holds column N.
  int mrow  = lane & 15;
  int khalf = (lane >> 4) << 1;               // 0 or 2
  long eL = base + mrow; if (eL >= NEDGES) eL = NEDGES - 1;
  const int rA = i0[eL];                      // A rows  from y[i0]
  const int rB = i1[eL];                      // B cols  from y[i1]

  v8f c = {};
  const float* ya = y + (size_t)rA * DIM + khalf;
  const float* yb = y + (size_t)rB * DIM + khalf;
#pragma unroll
  for (int k0 = 0; k0 < DIM; k0 += 4) {
    v2f a = *(const v2f*)(ya + k0);
    v2f b = *(const v2f*)(yb + k0);
    a *= rsD;
    c = __builtin_amdgcn_wmma_f32_16x16x4_f32(false, a, false, b,
                                              (short)0, c, false, false);
  }
  const float* ea = eigs + (size_t)rA * EIGD + khalf;
  const float* eb = eigs + (size_t)rB * EIGD + khalf;
#pragma unroll
  for (int k0 = 0; k0 < EIGD; k0 += 4) {
    v2f a = *(const v2f*)(ea + k0);
    v2f b = *(const v2f*)(eb + k0);
    a *= expl0;
    c = __builtin_amdgcn_wmma_f32_16x16x4_f32(false, a, false, b,
                                              (short)0, c, false, false);
  }

  // C layout: VGPR r, lanes 0-15 -> (M=r,   N=lane),
  //                  lanes 16-31 -> (M=r+8, N=lane-16).
  // Diagonal m: m<8 -> lane m comp m;  m>=8 -> lane m+16 comp m-8.
  int  m    = -1;
  float sv  = 0.0f;
  if (lane < 8)        { m = lane;      sv = c[lane]; }
  else if (lane >= 24) { m = lane - 16; sv = c[lane - 24]; }
  if (m >= 0) {
    long e = base + m;
    if (e < NEDGES) {
      int dst = i0[e];
      s0[e] = sv;
      atomicMax(&m0[dst], f2key(sv));
      float pv = pathw[pt[e]];
      s1[e] = pv;
      atomicMax(&m1[dst], f2key(pv));
    }
  }
}

// exp(s - m[dst]) in place + segment sums via f32 atomics.
__global__ void k_exp_sum(const int* __restrict__ i0,
                          float* __restrict__ s0, float* __restrict__ s1,
                          const unsigned* __restrict__ m0,
                          const unsigned* __restrict__ m1,
                          float* __restrict__ z0, float* __restrict__ z1) {
  size_t e = (size_t)blockIdx.x * blockDim.x + threadIdx.x;
  if (e >= NEDGES) return;
  int dst = i0[e];
  float e0 = __expf(s0[e] - key2f(m0[dst]));
  float e1 = __expf(s1[e] - key2f(m1[dst]));
  s0[e] = e0; s1[e] = e1;
  atomicAdd(&z0[dst], e0);
  atomicAdd(&z1[dst], e1);
}

// out[i0] += a * y[i1], one edge per wave, 4 floats per lane (f32 atomics; L2-resident).
__global__ void k_scatter(const float* __restrict__ y,
                          const int* __restrict__ i0, const int* __restrict__ i1,
                          const float* __restrict__ s0, const float* __restrict__ s1,
                          const float* __restrict__ z0, const float* __restrict__ z1,
                          float* __restrict__ xout) {
  int wid  = (int)((blockIdx.x * (size_t)blockDim.x + threadIdx.x) >> 5);
  int lane = threadIdx.x & 31;
  if (wid >= NEDGES) return;
  int dst = i0[wid], src = i1[wid];
  float a = 0.5f * (s0[wid] / (z0[dst] + 1e-16f) + s1[wid] / (z1[dst] + 1e-16f));
  float4 v = *(const float4*)(y + (size_t)src * DIM + lane * 4);
  float* o = xout + (size_t)dst * DIM + lane * 4;
  atomicAdd(o + 0, a * v.x);
  atomicAdd(o + 1, a * v.y);
  atomicAdd(o + 2, a * v.z);
  atomicAdd(o + 3, a * v.w);
}

// out += x / 3  (float4 streaming)
__global__ void k_accum(const float* __restrict__ x, float* __restrict__ out) {
  size_t i4 = (size_t)blockIdx.x * blockDim.x + threadIdx.x;
  const size_t tot4 = (size_t)NNODES * DIM / 4;
  if (i4 >= tot4) return;
  float4 v = ((const float4*)x)[i4];
  float4 o = ((float4*)out)[i4];
  o.x += v.x * (1.0f / 3.0f); o.y += v.y * (1.0f / 3.0f);
  o.z += v.z * (1.0f / 3.0f); o.w += v.w * (1.0f / 3.0f);
  ((float4*)out)[i4] = o;
}

// ---------- host ----------

extern "C" void kernel_launch(void* const* d_in, const int* in_sizes, int n_in,
                              void* d_out, int out_size, void* d_ws, size_t ws_size,
                              hipStream_t stream) {
  const float* user_emb = (const float*)d_in[0];   // [NUSERS, D]
  const float* item_emb = (const float*)d_in[1];   // [NITEMS, D]
  const float* eigs     = (const float*)d_in[2];   // [N, EIGD]
  const float* lambda0  = (const float*)d_in[3];   // [1]
  const float* pathw    = (const float*)d_in[4];   // [NPATH, 1]
  const int*   indices  = (const int*)d_in[5];     // [L, 2, E]
  const int*   ptype    = (const int*)d_in[6];     // [L, E]
  float*       out      = (float*)d_out;           // [N, D]

  const size_t ND = (size_t)NNODES * DIM;
  float* ws   = (float*)d_ws;
  float* bufA = ws;                    // N*D
  float* bufB = bufA + ND;             // N*D
  float* s0   = bufB + ND;             // E
  float* s1   = s0 + NEDGES;           // E
  unsigned* m0 = (unsigned*)(s1 + NEDGES);  // N (ordered keys)
  unsigned* m1 = m0 + NNODES;               // N
  float* z0   = (float*)(m1 + NNODES);      // N
  float* z1   = z0 + NNODES;                // N

  const int T = 256;
  const int gElem  = (int)((ND / 4 + T - 1) / T);        // 37500
  const int gRows  = (NNODES + 7) / 8;                   // 8 rows/block (8 waves)
  const int gEdgeW = ((NEDGES + 15) / 16 + 7) / 8;       // 16 edges/wave, 8 waves/block
  const int gEdgeT = (NEDGES + T - 1) / T;
  const int gScat  = (NEDGES + 7) / 8;                   // 1 edge/wave, 8 waves/block

  k_init<<<gElem, T, 0, stream>>>(user_emb, item_emb, bufA, out);

  float* xin  = bufA;
  float* xout = bufB;
  for (int l = 0; l < NLAYERS; ++l) {
    const int* i0 = indices + (size_t)l * 2 * NEDGES;
    const int* i1 = i0 + NEDGES;
    const int* pt = ptype + (size_t)l * NEDGES;

    k_layernorm<<<gRows, T, 0, stream>>>(xin);      // xin becomes y (in place)

    hipMemsetAsync(m0, 0, sizeof(unsigned) * NNODES, stream);  // key identity
    hipMemsetAsync(m1, 0, sizeof(unsigned) * NNODES, stream);
    hipMemsetAsync(z0, 0, sizeof(float) * NNODES, stream);
    hipMemsetAsync(z1, 0, sizeof(float) * NNODES, stream);
    hipMemsetAsync(xout, 0, sizeof(float) * ND, stream);

    k_edge_scores<<<gEdgeW, T, 0, stream>>>(xin, eigs, i0, i1, pt,
                                            lambda0, pathw, s0, s1, m0, m1);
    k_exp_sum<<<gEdgeT, T, 0, stream>>>(i0, s0, s1, m0, m1, z0, z1);
    k_scatter<<<gScat, T, 0, stream>>>(xin, i0, i1, s0, s1, z0, z1, xout);
    k_accum<<<gElem, T, 0, stream>>>(xout, out);

    float* t = xin; xin = xout; xout = t;
  }
}